// HadamardTest_52243982188997
// MI455X (gfx1250) — compile-verified
//
#include <hip/hip_runtime.h>

typedef float v2f __attribute__((ext_vector_type(2)));
typedef float v8f __attribute__((ext_vector_type(8)));

#define DIM     1024
#define NROWS   16384
#define NCLS    10
#define IMG     784
#define KSTEPS  (DIM / 4)   // 256 WMMA k-steps of K=4

// ---------------- Kernel 1: per-class inverse L2 norm of canon ----------------
__global__ __launch_bounds__(256) void ht_norm_kernel(const float* __restrict__ canon,
                                                      float* __restrict__ rnorm) {
    __shared__ float red[256];
    const int c = blockIdx.x;          // 0..9
    float s = 0.0f;
    for (int i = threadIdx.x; i < IMG; i += 256) {
        float v = canon[c * IMG + i];
        s += v * v;
    }
    red[threadIdx.x] = s;
    __syncthreads();
    for (int off = 128; off > 0; off >>= 1) {
        if (threadIdx.x < off) red[threadIdx.x] += red[threadIdx.x + off];
        __syncthreads();
    }
    if (threadIdx.x == 0) rnorm[c] = rsqrtf(red[0]);
}

// ---------------- Kernel 2: pack normalized ref into WMMA B lane order --------
// B layout for V_WMMA_F32_16X16X4_F32 (4x16 B-matrix, 2 VGPRs, wave32):
//   lane l = 16*half + n  holds  B[k = 2*half + j][n]  in vgpr j.
// Buffer layout: Bbuf[s*64 + l*2 + j] = refpad[class = l&15][dim = 4*s + 2*(l>>4) + j]
__global__ __launch_bounds__(256) void ht_packB_kernel(const float* __restrict__ canon,
                                                       const float* __restrict__ rnorm,
                                                       float* __restrict__ Bbuf) {
    const int o    = blockIdx.x * 256 + threadIdx.x;  // 0..16383
    const int s    = o >> 6;
    const int t    = o & 63;
    const int l    = t >> 1;
    const int j    = t & 1;
    const int c    = l & 15;
    const int half = l >> 4;
    const int d    = 4 * s + 2 * half + j;
    float v = 0.0f;
    if (c < NCLS && d < IMG) v = canon[c * IMG + d] * rnorm[c];
    Bbuf[o] = v;
}

// ---------------- Kernel 3: streamed fp32 WMMA swap-test GEMM -----------------
// One wave per 16-row tile of z; 4 waves (128 threads) per block.
__global__ __launch_bounds__(128) void ht_swaptest_kernel(const float* __restrict__ z_re,
                                                          const float* __restrict__ z_im,
                                                          const float* __restrict__ Bbuf,
                                                          float* __restrict__ out) {
    const int lane = threadIdx.x & 31;
    const int wave = threadIdx.x >> 5;
    const int tile = blockIdx.x * 4 + wave;
    const int base = tile * 16;
    const int m    = lane & 15;
    const int half = lane >> 4;

    // A-matrix per-lane stream: row (base+m), k = 4*s + 2*half + {0,1}
    const float* ar = z_re + (size_t)(base + m) * DIM + 2 * half;
    const float* ai = z_im + (size_t)(base + m) * DIM + 2 * half;
    // B per-lane stream: 64 consecutive floats per step (cache resident)
    const float* bp = Bbuf + lane * 2;

    v8f acc_re = {};
    v8f acc_im = {};

#pragma unroll 4
    for (int s = 0; s < KSTEPS; ++s) {
        v2f a0 = *(const v2f*)(ar + 4 * s);
        v2f a1 = *(const v2f*)(ai + 4 * s);
        v2f b  = *(const v2f*)(bp + 64 * s);
        // D = A*B + C, full fp32: v_wmma_f32_16x16x4_f32
        acc_re = __builtin_amdgcn_wmma_f32_16x16x4_f32(
            false, a0, false, b, (short)0, acc_re, false, false);
        acc_im = __builtin_amdgcn_wmma_f32_16x16x4_f32(
            false, a1, false, b, (short)0, acc_im, false, false);
    }

    // D layout: vgpr r, lanes 0-15 -> (M=r, N=lane); lanes 16-31 -> (M=8+r, N=lane-16)
    if (m < NCLS) {
#pragma unroll
        for (int r = 0; r < 8; ++r) {
            const float re = acc_re[r];
            const float im = acc_im[r];
            out[(size_t)(base + r + 8 * half) * NCLS + m] = re * re + im * im;
        }
    }
}

extern "C" void kernel_launch(void* const* d_in, const int* in_sizes, int n_in,
                              void* d_out, int out_size, void* d_ws, size_t ws_size,
                              hipStream_t stream) {
    const float* z_re  = (const float*)d_in[0];   // [16384, 1024]
    const float* z_im  = (const float*)d_in[1];   // [16384, 1024]
    const float* canon = (const float*)d_in[2];   // [10, 1, 28, 28]
    float* out   = (float*)d_out;                 // [16384, 10]
    float* Bbuf  = (float*)d_ws;                  // 16384 floats (64 KB)
    float* rnorm = Bbuf + 16 * DIM;               // 10 floats

    ht_norm_kernel<<<NCLS, 256, 0, stream>>>(canon, rnorm);
    ht_packB_kernel<<<(16 * DIM) / 256, 256, 0, stream>>>(canon, rnorm, Bbuf);
    ht_swaptest_kernel<<<NROWS / 64, 128, 0, stream>>>(z_re, z_im, Bbuf, out);
}